// GAT_46291157516305
// MI455X (gfx1250) — compile-verified
//
#include <hip/hip_runtime.h>
#include <math.h>

typedef __attribute__((ext_vector_type(2))) float v2f;
typedef __attribute__((ext_vector_type(8))) float v8f;

#define IN_CH 128
#define NEG_SLOPE 0.2f

// ---------------------------------------------------------------------------
// Kernel 1: h = x @ W  (out_channels = 1) via V_WMMA_F32_16X16X4_F32.
// Each wave32 handles 16 nodes. A (16x4) = W chunk broadcast down all 16 rows,
// B (4x16) = x[k, node] slice, so D[m][n] = h[node n] for every m.
// ISA layouts (§7.12.2):
//   A 16x4 : lanes 0-15 hold M=0..15 with VGPR{0,1}=K{0,1}; lanes 16-31 K{2,3}
//   B 4x16 : lanes 0-15 hold N=0..15 with VGPR{0,1}=K{0,1}; lanes 16-31 K{2,3}
//   D      : VGPR0, lanes 0-15 = (M=0, N=lane)  -> h[base+lane]
// EXEC must stay all-1s around the WMMA: per-wave-uniform exit, clamped loads.
// ---------------------------------------------------------------------------
__global__ __launch_bounds__(256) void gat_project_wmma(
    const float* __restrict__ x, const float* __restrict__ W,
    float* __restrict__ h, int n_nodes)
{
    const int lane = threadIdx.x & 31;
    const int wave = blockIdx.x * (blockDim.x >> 5) + (threadIdx.x >> 5);
    const int base = wave * 16;              // uniform within the wave
    if (base >= n_nodes) return;             // whole-wave exit (EXEC stays full)

    const int hi  = lane >> 4;               // 0: K pair {0,1}, 1: K pair {2,3}
    const int col = lane & 15;               // node column within the 16-tile
    int node = base + col;
    if (node >= n_nodes) node = n_nodes - 1; // clamp tail loads (redundant work)
    const float* __restrict__ xrow = x + (size_t)node * IN_CH;

    v8f acc = {};
#pragma unroll
    for (int k0 = 0; k0 < IN_CH; k0 += 4) {
        const int k = k0 + 2 * hi;
        v2f a; a.x = W[k];       a.y = W[k + 1];       // A[m][k] = W[k], all m
        v2f b; b.x = xrow[k];    b.y = xrow[k + 1];    // B[k][n] = x[node n][k]
        acc = __builtin_amdgcn_wmma_f32_16x16x4_f32(
            /*neg_a=*/false, a, /*neg_b=*/false, b,
            /*c_mod=*/(short)0, acc, /*reuse_a=*/false, /*reuse_b=*/false);
    }
    // Row M=0 of D: VGPR0 on lanes 0-15 holds h for nodes base..base+15.
    if (hi == 0 && (base + col) < n_nodes) h[base + col] = acc[0];
}

// ---------------------------------------------------------------------------
// Kernel 2: init reduction buffers. seg_max = -inf (bit pattern matters for
// the int-trick atomic max), num = denom = 0.
// ---------------------------------------------------------------------------
__global__ void gat_init(float* __restrict__ seg_max, float* __restrict__ num,
                         float* __restrict__ denom, int n)
{
    int i = blockIdx.x * blockDim.x + threadIdx.x;
    if (i < n) {
        seg_max[i] = __int_as_float(0xFF800000); // -inf
        num[i]     = 0.0f;
        denom[i]   = 0.0f;
    }
}

// Float atomic max via signed-max (val>=0) / unsigned-min (val<0).
// Correct for mixed-sign updates with -inf initial value.
__device__ __forceinline__ void atomicMaxF(float* addr, float val)
{
    if (val >= 0.0f)
        atomicMax((int*)addr, __float_as_int(val));
    else
        atomicMin((unsigned int*)addr, __float_as_uint(val));
}

__device__ __forceinline__ float leaky(float z)
{
    return z > 0.0f ? z : NEG_SLOPE * z;
}

// ---------------------------------------------------------------------------
// Kernel 3: edge pass 1 — per-destination max of leaky_relu logits.
// Edges e in [0,E) come from edge_index; e in [E, E+N) are self-loops.
// h (400 KB) stays resident in L2; the HBM traffic is the edge list.
// ---------------------------------------------------------------------------
__global__ void gat_edge_max(const int* __restrict__ src_idx,
                             const int* __restrict__ dst_idx,
                             const float* __restrict__ h,
                             const float* __restrict__ att_src,
                             const float* __restrict__ att_dst,
                             float* __restrict__ seg_max, int E, int N)
{
    int e = blockIdx.x * blockDim.x + threadIdx.x;
    if (e >= E + N) return;
    int s, d;
    if (e < E) { s = src_idx[e]; d = dst_idx[e]; }
    else       { s = d = e - E; }
    const float as = att_src[0], ad = att_dst[0];
    float logit = leaky(h[s] * as + h[d] * ad);
    atomicMaxF(&seg_max[d], logit);
}

// ---------------------------------------------------------------------------
// Kernel 4: edge pass 2 — accumulate Σe and Σe·h[src] per destination.
// Recomputes the logit (cheap VALU) instead of storing 26 MB of logits.
// ---------------------------------------------------------------------------
__global__ void gat_edge_acc(const int* __restrict__ src_idx,
                             const int* __restrict__ dst_idx,
                             const float* __restrict__ h,
                             const float* __restrict__ att_src,
                             const float* __restrict__ att_dst,
                             const float* __restrict__ seg_max,
                             float* __restrict__ num, float* __restrict__ denom,
                             int E, int N)
{
    int e = blockIdx.x * blockDim.x + threadIdx.x;
    if (e >= E + N) return;
    int s, d;
    if (e < E) { s = src_idx[e]; d = dst_idx[e]; }
    else       { s = d = e - E; }
    const float as = att_src[0], ad = att_dst[0];
    float hs = h[s];
    float logit = leaky(hs * as + h[d] * ad);
    float ex = expf(logit - seg_max[d]);
    atomicAdd(&denom[d], ex);
    atomicAdd(&num[d],   ex * hs);
}

// ---------------------------------------------------------------------------
// Kernel 5: out[i] = num[i] / (denom[i] + 1e-16) + bias
// ---------------------------------------------------------------------------
__global__ void gat_final(const float* __restrict__ num,
                          const float* __restrict__ denom,
                          const float* __restrict__ bias,
                          float* __restrict__ out, int n)
{
    int i = blockIdx.x * blockDim.x + threadIdx.x;
    if (i < n) out[i] = num[i] / (denom[i] + 1e-16f) + bias[0];
}

extern "C" void kernel_launch(void* const* d_in, const int* in_sizes, int n_in,
                              void* d_out, int out_size, void* d_ws, size_t ws_size,
                              hipStream_t stream)
{
    const float* x       = (const float*)d_in[0];
    const int*   edge    = (const int*)d_in[1];   // [2, E] row-major
    const float* W       = (const float*)d_in[2];
    const float* att_src = (const float*)d_in[3];
    const float* att_dst = (const float*)d_in[4];
    const float* bias    = (const float*)d_in[5];
    float*       out     = (float*)d_out;

    const int N = in_sizes[0] / IN_CH;
    const int E = in_sizes[1] / 2;

    // Workspace layout: h | seg_max | denom | num  (4*N floats = 1.6 MB)
    float* h       = (float*)d_ws;
    float* seg_max = h + N;
    float* denom   = seg_max + N;
    float* num     = denom + N;

    // 1) projection (WMMA): 256 threads = 8 waves = 128 nodes per block
    const int nodes_per_block = 16 * 8;
    gat_project_wmma<<<(N + nodes_per_block - 1) / nodes_per_block, 256, 0, stream>>>(
        x, W, h, N);

    // 2) init reductions
    gat_init<<<(N + 255) / 256, 256, 0, stream>>>(seg_max, num, denom, N);

    // 3) segment max over E real edges + N self-loops
    const int total = E + N;
    gat_edge_max<<<(total + 255) / 256, 256, 0, stream>>>(
        edge, edge + E, h, att_src, att_dst, seg_max, E, N);

    // 4) accumulate numerator/denominator
    gat_edge_acc<<<(total + 255) / 256, 256, 0, stream>>>(
        edge, edge + E, h, att_src, att_dst, seg_max, num, denom, E, N);

    // 5) finalize
    gat_final<<<(N + 255) / 256, 256, 0, stream>>>(num, denom, bias, out, N);
}